// SimilarityTripletLoss_40604620816673
// MI455X (gfx1250) — compile-verified
//
#include <hip/hip_runtime.h>
#include <hip/hip_bf16.h>

// Problem dims (fixed by the reference)
#define BB 8
#define CC 512
#define FH 64
#define FW 64
#define FHW (FH * FW)          // 4096
#define NCELLS (BB * FHW)      // 32768
#define MARGIN 12.0f

typedef __attribute__((ext_vector_type(2))) float v2f;
typedef __attribute__((ext_vector_type(8))) float v8f;

// ---------------------------------------------------------------------------
// Kernel 0: zero the two global accumulators (loss sum, valid-cell count)
// ---------------------------------------------------------------------------
__global__ void stl_init_acc(float* acc) {
    acc[0] = 0.0f;
    acc[1] = 0.0f;
}

// ---------------------------------------------------------------------------
// Kernel 1: LDS-tiled transpose (B, C, FHW) -> (B, FHW, C)
// grid: (FHW/32, C/32, B), block: (32, 8)
// ---------------------------------------------------------------------------
__global__ __launch_bounds__(256) void stl_transpose(const float* __restrict__ in,
                                                     float* __restrict__ out) {
    __shared__ float tile[32][33];
    const int b  = blockIdx.z;
    const int c0 = blockIdx.y * 32;
    const int p0 = blockIdx.x * 32;
    const int tx = threadIdx.x;  // 0..31 (pixel within tile on load)
    const int ty = threadIdx.y;  // 0..7

#pragma unroll
    for (int i = 0; i < 32; i += 8) {
        tile[ty + i][tx] =
            in[((size_t)b * CC + (c0 + ty + i)) * FHW + (p0 + tx)];
    }
    __syncthreads();
#pragma unroll
    for (int i = 0; i < 32; i += 8) {
        out[((size_t)b * FHW + (p0 + ty + i)) * CC + (c0 + tx)] = tile[tx][ty + i];
    }
}

// ---------------------------------------------------------------------------
// Kernel 2: main triplet kernel.
// One wave32 handles 4 cells = 16 (anchor, gathered) pairs.
// Squared distances computed as diag( diff * diff^T ) via
// V_WMMA_F32_16X16X4_F32, accumulated over K = 512 in steps of 4.
//
// A-matrix 16x4 f32 layout (ISA 7.12.2): lane L holds rows M = L%16,
// K = {2*(L/16), 2*(L/16)+1} in VGPRs {0,1}. The B-matrix 4x16 layout is the
// mirrored transpose, so feeding the SAME register pair as A and B gives
// D[m][n] = sum_k diff[m][k] * diff[n][k]; the diagonal is what we need.
//
// TRANSPOSED=true : vectors are contiguous (stride 1, base = pixel*CC)
// TRANSPOSED=false: vectors strided by FHW in the original layout
// ---------------------------------------------------------------------------
template <bool TRANSPOSED>
__global__ __launch_bounds__(256) void stl_triplet(const float* __restrict__ sk,
                                                   const float* __restrict__ rf,
                                                   const int* __restrict__ pos_idx,
                                                   const int* __restrict__ neg_idx,
                                                   const float* __restrict__ pair_w,
                                                   float* __restrict__ acc) {
    const int lane = threadIdx.x & 31;
    const int wave = threadIdx.x >> 5;         // 0..7
    const int p    = lane & 15;                // pair row 0..15
    const int hi   = lane >> 4;                // 0/1 (K half)

    const int cellBase = blockIdx.x * 32 + wave * 4;   // 4 cells per wave
    const int cell     = cellBase + (p >> 2);          // this lane's cell
    const int vec      = p & 3;                        // 0,1=pos ; 2,3=neg

    const int b  = cell >> 12;                 // / FHW
    const int hw = cell & (FHW - 1);

    // gathered pixel (y, x) into the featured map
    const int* idx = (vec < 2) ? pos_idx : neg_idx;
    const int  s   = vec & 1;
    const int  py  = idx[((size_t)cell * 2 + s) * 2 + 0];
    const int  px  = idx[((size_t)cell * 2 + s) * 2 + 1];
    const int  gl  = py * FW + px;             // pixel within (b, FHW)

    const float* aPtr;
    const float* vPtr;
    size_t stride;
    if (TRANSPOSED) {
        aPtr   = rf + (size_t)cell * CC;
        vPtr   = sk + ((size_t)b * FHW + gl) * CC;
        stride = 1;
    } else {
        aPtr   = rf + (size_t)b * CC * FHW + hw;
        vPtr   = sk + (size_t)b * CC * FHW + gl;
        stride = FHW;
    }
    // lane covers K positions {k0 + 2*hi, k0 + 2*hi + 1}
    const float* ap = aPtr + (size_t)(2 * hi) * stride;
    const float* vp = vPtr + (size_t)(2 * hi) * stride;

    v8f accv = {0.f, 0.f, 0.f, 0.f, 0.f, 0.f, 0.f, 0.f};

#pragma unroll 4
    for (int k0 = 0; k0 < CC; k0 += 4) {
        const float a0 = ap[0];
        const float a1 = ap[stride];
        const float w0 = vp[0];
        const float w1 = vp[stride];
        v2f frag;
        frag.x = a0 - w0;
        frag.y = a1 - w1;
        // D = A*B + C with A == B (register-aliased) -> Gram of diffs
        accv = __builtin_amdgcn_wmma_f32_16x16x4_f32(
            /*neg_a=*/false, frag, /*neg_b=*/false, frag,
            /*c_mod=*/(short)0, accv, /*reuse_a=*/false, /*reuse_b=*/false);
        ap += 4 * stride;
        vp += 4 * stride;
    }

    // Extract diagonal: pair q is read from lane q (q<8, VGPR q) or lane q+8
    // (q>=8, VGPR q-8).
    const int rsel = (lane < 8) ? lane : ((lane >= 16 && lane < 24) ? (lane - 16) : 0);
    float d = accv[0];
#pragma unroll
    for (int r = 1; r < 8; ++r) d = (rsel == r) ? accv[r] : d;

    // Route the 4 distances of cell (lane&3) to lanes 0..3 of the wave.
    const int cL = lane & 3;
    const int q0 = cL * 4 + 0, q1 = cL * 4 + 1, q2 = cL * 4 + 2, q3 = cL * 4 + 3;
    const float dp0 = __shfl(d, (q0 < 8) ? q0 : q0 + 8, 32);
    const float dp1 = __shfl(d, (q1 < 8) ? q1 : q1 + 8, 32);
    const float dn0 = __shfl(d, (q2 < 8) ? q2 : q2 + 8, 32);
    const float dn1 = __shfl(d, (q3 < 8) ? q3 : q3 + 8, 32);

    float loss = 0.0f, validf = 0.0f;
    if (lane < 4) {
        const int cellG = cellBase + lane;
        const float* w  = pair_w + (size_t)cellG * 4;
        const float w00 = w[0], w01 = w[1], w10 = w[2], w11 = w[3];
        loss = fmaxf(dp0 - dn0 + MARGIN, 0.0f) * w00 +
               fmaxf(dp0 - dn1 + MARGIN, 0.0f) * w01 +
               fmaxf(dp1 - dn0 + MARGIN, 0.0f) * w10 +
               fmaxf(dp1 - dn1 + MARGIN, 0.0f) * w11;
        validf = ((w00 + w01 + w10 + w11) > 0.0f) ? 1.0f : 0.0f;
    }

    // wave32 butterfly reduction
#pragma unroll
    for (int off = 16; off > 0; off >>= 1) {
        loss   += __shfl_xor(loss, off, 32);
        validf += __shfl_xor(validf, off, 32);
    }

    __shared__ float sl[8], sv[8];
    if (lane == 0) { sl[wave] = loss; sv[wave] = validf; }
    __syncthreads();
    if (threadIdx.x == 0) {
        float L = 0.0f, V = 0.0f;
#pragma unroll
        for (int i = 0; i < 8; ++i) { L += sl[i]; V += sv[i]; }
        atomicAdd(&acc[0], L);
        atomicAdd(&acc[1], V);
    }
}

// ---------------------------------------------------------------------------
// Kernel 3: finalize   out = sum / (1e-6 + n_valid)
// ---------------------------------------------------------------------------
__global__ void stl_finalize(const float* __restrict__ acc, float* __restrict__ out) {
    out[0] = acc[0] / (1e-6f + acc[1]);
}

// ---------------------------------------------------------------------------
extern "C" void kernel_launch(void* const* d_in, const int* in_sizes, int n_in,
                              void* d_out, int out_size, void* d_ws, size_t ws_size,
                              hipStream_t stream) {
    const float* sketch  = (const float*)d_in[0];  // (B,C,FH,FW)
    const float* ref     = (const float*)d_in[1];  // (B,C,FH,FW)
    // d_in[2] = G (unused by the math; indices are precomputed)
    const int*   pos_idx = (const int*)d_in[3];    // (B,FH,FW,2,2)
    const int*   neg_idx = (const int*)d_in[4];    // (B,FH,FW,2,2)
    const float* pair_w  = (const float*)d_in[5];  // (B,FH,FW,2,2)
    float*       out     = (float*)d_out;

    float* acc = (float*)d_ws;                     // 2 accumulators
    const size_t tOff     = 256;                   // keep arrays aligned past acc
    const size_t arrBytes = (size_t)BB * CC * FHW * sizeof(float);  // 64 MB
    const bool   useT     = ws_size >= tOff + 2 * arrBytes;

    stl_init_acc<<<1, 1, 0, stream>>>(acc);

    const dim3 mainGrid(NCELLS / 32);              // 32 cells per 256-thread block
    const dim3 mainBlock(256);

    if (useT) {
        float* skT  = (float*)((char*)d_ws + tOff);
        float* refT = (float*)((char*)d_ws + tOff + arrBytes);
        const dim3 tg(FHW / 32, CC / 32, BB);
        const dim3 tb(32, 8);
        stl_transpose<<<tg, tb, 0, stream>>>(sketch, skT);
        stl_transpose<<<tg, tb, 0, stream>>>(ref, refT);
        stl_triplet<true><<<mainGrid, mainBlock, 0, stream>>>(
            skT, refT, pos_idx, neg_idx, pair_w, acc);
    } else {
        stl_triplet<false><<<mainGrid, mainBlock, 0, stream>>>(
            sketch, ref, pos_idx, neg_idx, pair_w, acc);
    }

    stl_finalize<<<1, 1, 0, stream>>>(acc, out);
}